// LBEADS_NET_3444563771731
// MI455X (gfx1250) — compile-verified
//
#include <hip/hip_runtime.h>
#include <math.h>

#define NN 1024
#define BB 8
#define R_ASYM 6.0f
#define EPS_0 1e-6f
#define EPS_1 1e-6f

typedef float v2f __attribute__((ext_vector_type(2)));
typedef float v4f __attribute__((ext_vector_type(4)));
typedef float v8f __attribute__((ext_vector_type(8)));

// ---------------------------------------------------------------------------
// ATb = A^T * b_vec, b_vec = (1-R)/2 * ones.  A is tridiagonal -> column sums.
// ---------------------------------------------------------------------------
__global__ void k_atb(const float* __restrict__ A, float* __restrict__ atb) {
    int i = blockIdx.x * blockDim.x + threadIdx.x;
    if (i >= NN) return;
    float s = A[(size_t)i * NN + i];
    if (i > 0)      s += A[(size_t)(i - 1) * NN + i];
    if (i < NN - 1) s += A[(size_t)(i + 1) * NN + i];
    atb[i] = ((1.0f - R_ASYM) * 0.5f) * s;   // -2.5 * colsum
}

// ---------------------------------------------------------------------------
// z0 = A^{-1} y  (Thomas algorithm, A tridiagonal; one thread per batch row)
// ---------------------------------------------------------------------------
__global__ void k_thomas(const float* __restrict__ A, const float* __restrict__ y,
                         float* __restrict__ cpr, float* __restrict__ z0) {
    int b = threadIdx.x;
    if (b >= BB) return;
    float*       cp = cpr + (size_t)b * NN;
    float*       z  = z0  + (size_t)b * NN;
    const float* yb = y   + (size_t)b * NN;

    float ad = A[0], au = A[1];
    float c  = au / ad;
    float dp = yb[0] / ad;
    cp[0] = c; z[0] = dp;
    for (int i = 1; i < NN; ++i) {
        float al = A[(size_t)i * NN + (i - 1)];
        ad = A[(size_t)i * NN + i];
        au = (i < NN - 1) ? A[(size_t)i * NN + i + 1] : 0.0f;
        float m = ad - al * c;
        c  = au / m;
        dp = (yb[i] - al * dp) / m;
        cp[i] = c; z[i] = dp;
    }
    for (int i = NN - 2; i >= 0; --i) z[i] = z[i] - cp[i] * z[i + 1];
}

__global__ void k_copy(const float* __restrict__ src, float* __restrict__ dst) {
    int i = blockIdx.x * blockDim.x + threadIdx.x;
    if (i < BB * NN) dst[i] = src[i];
}

// Zero-pad [8,K] -> [16,K] so the WMMA GEMM loop is branch-free.
__global__ void k_pad16(const float* __restrict__ src, float* __restrict__ dst) {
    int i = blockIdx.x * blockDim.x + threadIdx.x;
    if (i >= 16 * NN) return;
    int r = i >> 10;
    dst[i] = (r < BB) ? src[i] : 0.0f;
}

// ---------------------------------------------------------------------------
// Skinny GEMM via V_WMMA_F32_16X16X4_F32:
//   OUT[8, Mout] = Xpad[16, K] * W^T   with W row-major [Mout, K], 16 | Mout.
// One wave per 16-column tile.  Per 8-wide K block we remap K so each lane's
// data for two consecutive WMMAs is one contiguous b128:
//   half-0 lanes carry K = k0..k0+3, half-1 lanes carry K = k0+4..k0+7;
//   WMMA #1 uses lane floats {0,1}, WMMA #2 uses {2,3}.  A and B use the
//   identical K mapping, so the K-sum is unchanged.
// ---------------------------------------------------------------------------
__global__ void k_gemm_wmma(const float* __restrict__ Xpad, const float* __restrict__ W,
                            float* __restrict__ OUT, int K, int Mout) {
    int lane = threadIdx.x;      // 0..31
    int half = lane >> 4;
    int crow = lane & 15;
    int col  = blockIdx.x * 16 + crow;

    const float* xrow = Xpad + (size_t)crow * K + 4 * half;
    const float* wrow = W    + (size_t)col  * K + 4 * half;
    v8f acc = {};

#pragma unroll 4
    for (int k0 = 0; k0 < K; k0 += 8) {
        __builtin_prefetch(wrow + k0 + 128, 0, 1);   // stream W ahead (global_prefetch)
        v4f a4 = *(const v4f*)(xrow + k0);
        v4f b4 = *(const v4f*)(wrow + k0);
        v2f a0 = __builtin_shufflevector(a4, a4, 0, 1);
        v2f a1 = __builtin_shufflevector(a4, a4, 2, 3);
        v2f b0 = __builtin_shufflevector(b4, b4, 0, 1);
        v2f b1 = __builtin_shufflevector(b4, b4, 2, 3);
        acc = __builtin_amdgcn_wmma_f32_16x16x4_f32(false, a0, false, b0,
                                                    (short)0, acc, false, false);
        acc = __builtin_amdgcn_wmma_f32_16x16x4_f32(false, a1, false, b1,
                                                    (short)0, acc, false, false);
    }

    if (half == 0) {             // lanes 0..15 hold M = 0..7 in acc[0..7]
#pragma unroll
        for (int r = 0; r < 8; ++r)
            OUT[(size_t)r * Mout + col] = acc[r];
    }
}

// ---------------------------------------------------------------------------
// Per-layer elementwise: IRLS weights u (1st diff), v (2nd diff),
// gamma' = 2*lam0*(1+R)/4 / max(|x|,eps), and d_vec = BTBz - lam0*ATb.
// ---------------------------------------------------------------------------
__global__ void k_weights(const float* __restrict__ x, const float* __restrict__ btbz,
                          const float* __restrict__ atb, const float* __restrict__ loglams,
                          int layer, float* __restrict__ uarr, float* __restrict__ varr,
                          float* __restrict__ gam, float* __restrict__ dvec) {
    int idx = blockIdx.x * blockDim.x + threadIdx.x;
    if (idx >= BB * NN) return;
    int b = idx >> 10, i = idx & (NN - 1);
    float lam0 = expf(loglams[layer * 3 + 0]);
    float lam1 = expf(loglams[layer * 3 + 1]);
    float lam2 = expf(loglams[layer * 3 + 2]);
    const float* xb = x + (size_t)b * NN;
    float xi = xb[i];
    gam[idx] = 2.0f * lam0 * ((1.0f + R_ASYM) * 0.25f) / fmaxf(fabsf(xi), EPS_0);
    if (i < NN - 1) {
        float d1 = xb[i + 1] - xi;
        uarr[idx] = lam1 / (fabsf(d1) + EPS_1);
    }
    if (i < NN - 2) {
        float d2 = xi - 2.0f * xb[i + 1] + xb[i + 2];
        varr[idx] = lam2 / (fabsf(d2) + EPS_1);
    }
    dvec[idx] = btbz[idx] - lam0 * atb[i];
}

// M = 2*lam0*diag(gamma) + D^T Lambda D  is pentadiagonal; evaluate entries.
__device__ __forceinline__ float Uat(const float* u, int m) {
    return (m >= 0 && m < NN - 1) ? u[m] : 0.0f;
}
__device__ __forceinline__ float Vat(const float* v, int m) {
    return (m >= 0 && m < NN - 2) ? v[m] : 0.0f;
}
__device__ __forceinline__ float Mval(const float* u, const float* v, const float* g,
                                      int p, int q) {
    if (p < 0 || q < 0 || p >= NN || q >= NN) return 0.0f;
    int dd = q - p;
    if (dd < 0) { int t = p; p = q; q = t; dd = -dd; }
    if (dd > 2) return 0.0f;
    if (dd == 0)
        return Uat(u, p - 1) + Uat(u, p) + Vat(v, p) + 4.0f * Vat(v, p - 1) +
               Vat(v, p - 2) + g[p];
    if (dd == 1)
        return -Uat(u, p) - 2.0f * Vat(v, p) - 2.0f * Vat(v, p - 1);
    return Vat(v, p);
}

// ---------------------------------------------------------------------------
// lhs = BTB + A^T M A, banded bw=4 (A tridiagonal Toeplitz).
// (A^T M A)[i,j] = sum_{s,t in {-1,0,1}} a_s a_t M[i-s, j-t].
// Store lower band: band[b][r][d] = lhs[r][r-d], d=0..4.
// ---------------------------------------------------------------------------
__global__ void k_build(const float* __restrict__ A, const float* __restrict__ BTB,
                        const float* __restrict__ uarr, const float* __restrict__ varr,
                        const float* __restrict__ gam, float* __restrict__ band) {
    int idx = blockIdx.x * blockDim.x + threadIdx.x;
    if (idx >= BB * NN) return;
    int b = idx >> 10, i = idx & (NN - 1);
    float as_[3] = { A[NN], A[0], A[1] };   // a_{-1}, a_0, a_{+1}
    const float* ub = uarr + (size_t)b * NN;
    const float* vb = varr + (size_t)b * NN;
    const float* gb = gam  + (size_t)b * NN;

#pragma unroll
    for (int o = 0; o <= 4; ++o) {
        int r = i + o;
        if (r >= NN) break;
        float val = BTB[(size_t)i * NN + r];
#pragma unroll
        for (int s = -1; s <= 1; ++s)
#pragma unroll
            for (int t = -1; t <= 1; ++t)
                val += as_[s + 1] * as_[t + 1] * Mval(ub, vb, gb, i - s, r - t);
        band[((size_t)b * NN + r) * 5 + o] = val;
    }
}

// ---------------------------------------------------------------------------
// Banded Cholesky (bandwidth 4) + triangular solves, one block per sample.
// Band staged into LDS so the serial chain runs at LDS latency.
// ---------------------------------------------------------------------------
__global__ void k_chol(const float* __restrict__ band, const float* __restrict__ dvec,
                       float* __restrict__ zbuf) {
    __shared__ float sb[NN * 5];
    __shared__ float sz[NN];
    __shared__ float sd[NN];
    int b = blockIdx.x, tid = threadIdx.x;
    for (int j = tid; j < NN * 5; j += blockDim.x)
        sb[j] = band[(size_t)b * NN * 5 + j];
    for (int j = tid; j < NN; j += blockDim.x)
        sd[j] = dvec[(size_t)b * NN + j];
    __syncthreads();

    if (tid == 0) {
        // Row-oriented band Cholesky: L[i][k] for k = i-4..i
        for (int i = 0; i < NN; ++i) {
            int lo = (i - 4 < 0) ? 0 : i - 4;
            for (int k = lo; k < i; ++k) {
                float s = sb[i * 5 + (i - k)];
                for (int m = lo; m < k; ++m)
                    s -= sb[i * 5 + (i - m)] * sb[k * 5 + (k - m)];
                sb[i * 5 + (i - k)] = s / sb[k * 5];
            }
            float s = sb[i * 5];
            for (int m = lo; m < i; ++m) {
                float l = sb[i * 5 + (i - m)];
                s -= l * l;
            }
            sb[i * 5] = sqrtf(fmaxf(s, 1e-20f));
        }
        // Forward:  L y = d
        for (int i = 0; i < NN; ++i) {
            float s = sd[i];
            int dmax = (i < 4) ? i : 4;
            for (int d = 1; d <= dmax; ++d) s -= sb[i * 5 + d] * sz[i - d];
            sz[i] = s / sb[i * 5];
        }
        // Backward: L^T z = y
        for (int i = NN - 1; i >= 0; --i) {
            float s = sz[i];
            for (int d = 1; d <= 4; ++d)
                if (i + d < NN) s -= sb[(i + d) * 5 + d] * sz[i + d];
            sz[i] = s / sb[i * 5];
        }
    }
    __syncthreads();
    for (int j = tid; j < NN; j += blockDim.x)
        zbuf[(size_t)b * NN + j] = sz[j];
}

__global__ void k_final(const float* __restrict__ x, const float* __restrict__ y,
                        float* __restrict__ out) {
    int i = blockIdx.x * blockDim.x + threadIdx.x;
    if (i >= BB * NN) return;
    float xv = x[i];
    out[i]           = xv;        // x
    out[BB * NN + i] = y[i] - xv; // f = y - x
}

// ---------------------------------------------------------------------------
extern "C" void kernel_launch(void* const* d_in, const int* in_sizes, int n_in,
                              void* d_out, int out_size, void* d_ws, size_t ws_size,
                              hipStream_t stream) {
    const float* y       = (const float*)d_in[0];
    const float* A       = (const float*)d_in[1];
    // d_in[2] = Dm: stencil applied analytically, not needed as dense matrix
    const float* BTB     = (const float*)d_in[3];
    const float* loglams = (const float*)d_in[4];
    float* out = (float*)d_out;

    float* w    = (float*)d_ws;
    float* cpr  = w;  w += BB * NN;       // Thomas scratch
    float* z0   = w;  w += BB * NN;       // A^{-1} y
    float* btbz = w;  w += BB * NN;       // BTB z0
    float* atb  = w;  w += NN;            // A^T b
    float* xbuf = w;  w += BB * NN;       // running x
    float* dvec = w;  w += BB * NN;
    float* uarr = w;  w += BB * NN;
    float* varr = w;  w += BB * NN;
    float* gam  = w;  w += BB * NN;
    float* zbuf = w;  w += BB * NN;
    float* xpad = w;  w += 16 * NN;       // zero-padded GEMM input [16,N]
    float* band = w;  w += (size_t)BB * NN * 5;

    k_atb   <<<(NN + 255) / 256, 256, 0, stream>>>(A, atb);
    k_thomas<<<1, 32, 0, stream>>>(A, y, cpr, z0);
    k_pad16 <<<(16 * NN + 255) / 256, 256, 0, stream>>>(z0, xpad);
    k_gemm_wmma<<<NN / 16, 32, 0, stream>>>(xpad, BTB, btbz, NN, NN);   // BTB_z
    k_copy  <<<(BB * NN + 255) / 256, 256, 0, stream>>>(y, xbuf);       // x = y

    for (int layer = 0; layer < 3; ++layer) {
        k_weights<<<(BB * NN + 255) / 256, 256, 0, stream>>>(
            xbuf, btbz, atb, loglams, layer, uarr, varr, gam, dvec);
        k_build  <<<(BB * NN + 255) / 256, 256, 0, stream>>>(
            A, BTB, uarr, varr, gam, band);
        k_chol   <<<BB, 32, 0, stream>>>(band, dvec, zbuf);
        k_pad16  <<<(16 * NN + 255) / 256, 256, 0, stream>>>(zbuf, xpad);
        k_gemm_wmma<<<NN / 16, 32, 0, stream>>>(xpad, A, xbuf, NN, NN);  // x = A z
    }

    k_final<<<(BB * NN + 255) / 256, 256, 0, stream>>>(xbuf, y, out);
}